// custom_loss_9981503995978
// MI455X (gfx1250) — compile-verified
//
#include <hip/hip_runtime.h>

typedef float v2f __attribute__((ext_vector_type(2)));
typedef float v4f __attribute__((ext_vector_type(4)));
typedef float v8f __attribute__((ext_vector_type(8)));

#define ROWLEN 961
#define ROWS_PER_WAVE 16
#define WTAB_PITCH 976   // 976*4 = 3904 bytes, 16B-aligned row pitch

#define WMMA_F32_4(A, B, C) \
    __builtin_amdgcn_wmma_f32_16x16x4_f32(false, (A), false, (B), (short)0, (C), false, false)

__global__ __launch_bounds__(256) void heatmap_softargmax_dist(
    const float* __restrict__ pred, const float* __restrict__ gt,
    float* __restrict__ out, float inv_rows)
{
    // Weight table: row 0 = x(e) = e%31, row 1 = y(e) = e/31. Built once per block.
    __shared__ __align__(16) float wtab[2][WTAB_PITCH];
    for (int i = threadIdx.x; i < ROWLEN; i += 256) {
        int y = i / 31;
        int x = i - y * 31;
        wtab[0][i] = (float)x;
        wtab[1][i] = (float)y;
    }
    __syncthreads();

    const int lane      = threadIdx.x & 31;
    const int waveInBlk = threadIdx.x >> 5;
    const int wave      = blockIdx.x * 8 + waveInBlk;
    const int n   = lane & 15;   // A: tile row;  B: output column
    const int hi  = lane >> 4;   // lane half supplies K0,K1 (hi=0) or K2,K3 (hi=1)
    const int row = wave * ROWS_PER_WAVE + n;

    const float* prow = pred + (size_t)row * ROWLEN;
    const float* grow = gt   + (size_t)row * ROWLEN;
    // Accumulator columns >= 2 are never read -> lanes n>=2 may carry arbitrary
    // finite weights; just pick x-row (n==0) or y-row (else). No masking needed.
    const float* wrow = wtab[(n == 0) ? 0 : 1];
    const int offB = hi * 4;     // element offset within the 8-wide K window

    v8f accP = {};
    v8f accG = {};

    // Software-pipelined: prefetch window w+1 while WMMAs consume window w.
    v4f p, g, wv;
    __builtin_memcpy(&p, prow + offB, 16);       // 4B-aligned -> global_load_b128
    __builtin_memcpy(&g, grow + offB, 16);
    wv = *(const v4f*)(wrow + offB);             // 16B-aligned -> ds_load_b128

    #pragma unroll 7
    for (int w = 0; w < 119; ++w) {
        const int cn = (w + 1) * 8;
        v4f pn, gn, wn;
        __builtin_memcpy(&pn, prow + cn + offB, 16);
        __builtin_memcpy(&gn, grow + cn + offB, 16);
        wn = *(const v4f*)(wrow + cn + offB);

        v2f b0 = { wv.x, wv.y }, b1 = { wv.z, wv.w };
        v2f a0 = { p.x, p.y },   a1 = { p.z, p.w };
        v2f c0 = { g.x, g.y },   c1 = { g.z, g.w };
        accP = WMMA_F32_4(a0, b0, accP);
        accP = WMMA_F32_4(a1, b1, accP);
        accG = WMMA_F32_4(c0, b0, accG);
        accG = WMMA_F32_4(c1, b1, accG);

        p = pn; g = gn; wv = wn;
    }
    // Drain: window 119 (elements 952..959)
    {
        v2f b0 = { wv.x, wv.y }, b1 = { wv.z, wv.w };
        v2f a0 = { p.x, p.y },   a1 = { p.z, p.w };
        v2f c0 = { g.x, g.y },   c1 = { g.z, g.w };
        accP = WMMA_F32_4(a0, b0, accP);
        accP = WMMA_F32_4(a1, b1, accP);
        accG = WMMA_F32_4(c0, b0, accG);
        accG = WMMA_F32_4(c1, b1, accG);
    }

    // Tail element 960: x = y = 30, only K0 carries data; hi lanes feed zero A.
    {
        float pt = hi ? 0.0f : prow[960];
        float qt = hi ? 0.0f : grow[960];
        v2f bt = { 30.0f, 0.0f };
        v2f at = { pt, 0.0f };
        v2f ag = { qt, 0.0f };
        accP = WMMA_F32_4(at, bt, accP);
        accG = WMMA_F32_4(ag, bt, accG);
    }

    // col0 = sum(v*x), col1 = sum(v*y); px = 31*sum/961 -> scale s = 31/961
    const float s = 31.0f / 961.0f;
    float partial = 0.0f;
    #pragma unroll
    for (int m = 0; m < 8; ++m) {
        float d = (accP[m] - accG[m]) * s;      // lane0/16: dx of pair m / m+8; lane1/17: dy
        float o = __shfl_xor(d, 1);             // pair up dx with dy
        partial += sqrtf(d * d + o * o) * 0.5f;
    }
    // Only lanes 0 and 16 hold meaningful partials; lanes 1/17 are duplicates,
    // lanes >=2 hold garbage that is never read below.
    partial += __shfl_xor(partial, 16);         // lane 0: all 16 dists of this wave

    __shared__ float ws[8];
    if (lane == 0) ws[waveInBlk] = partial;
    __syncthreads();
    if (threadIdx.x == 0) {
        float bs = 0.0f;
        #pragma unroll
        for (int i = 0; i < 8; ++i) bs += ws[i];
        atomicAdd(out, bs * inv_rows);
    }
}

extern "C" void kernel_launch(void* const* d_in, const int* in_sizes, int n_in,
                              void* d_out, int out_size, void* d_ws, size_t ws_size,
                              hipStream_t stream) {
    const float* pred = (const float*)d_in[0];
    const float* gt   = (const float*)d_in[1];
    float* out        = (float*)d_out;

    const int nrows  = in_sizes[0] / ROWLEN;        // B*J = 65536
    const int nwaves = nrows / ROWS_PER_WAVE;       // 4096
    const int blocks = nwaves / 8;                  // 512 blocks of 8 waves

    hipMemsetAsync(out, 0, sizeof(float), stream);  // harness poisons d_out
    heatmap_softargmax_dist<<<blocks, 256, 0, stream>>>(pred, gt, out, 1.0f / (float)nrows);
}